// Net_17489106829462
// MI455X (gfx1250) — compile-verified
//
#include <hip/hip_runtime.h>
#include <hip/hip_bf16.h>
#include <stdint.h>

#define N_NODES 100000
#define N_EDGES 3200000
#define F_IN    128
#define F_HID   32
#define F_OUT   40
#define M_TILES (N_NODES / 16)   // 6250, exact

typedef float v8f __attribute__((ext_vector_type(8)));
typedef float v2f __attribute__((ext_vector_type(2)));

// ---------------------------------------------------------------------------
// GEMM1: h1[100000x32] = x[100000x128] @ W1[128x32], f32 WMMA 16x16x4.
// One wave per 16x16 output tile; 2 N-tiles, 6250 M-tiles.
// A layout: lane m=L%16, group g=L/16 holds K = 4*kt + 2g + {0,1}.
// B layout: lane n=L%16, group g holds rows K = 4*kt + 2g + {0,1}.
// D layout: lane n=L%16; VGPR v holds row m = v + 8g.
// ---------------------------------------------------------------------------
__global__ void gcn_gemm1(const float* __restrict__ x,
                          const float* __restrict__ W1,
                          float* __restrict__ h1)
{
    const int wave = (blockIdx.x * blockDim.x + threadIdx.x) >> 5;
    const int lane = threadIdx.x & 31;
    if (wave >= M_TILES * 2) return;
    const int mt = wave >> 1;
    const int nt = wave & 1;
    const int mi = lane & 15;
    const int g  = lane >> 4;
    const int n  = nt * 16 + mi;

    const float* __restrict__ xrow = x + (size_t)(mt * 16 + mi) * F_IN;
    v8f c = {};
#pragma unroll 4
    for (int kt = 0; kt < F_IN / 4; ++kt) {
        const int k0 = kt * 4 + 2 * g;
        v2f a, b;
        a.x = xrow[k0];
        a.y = xrow[k0 + 1];
        b.x = W1[(size_t)k0 * F_HID + n];
        b.y = W1[(size_t)(k0 + 1) * F_HID + n];
        c = __builtin_amdgcn_wmma_f32_16x16x4_f32(false, a, false, b,
                                                  (short)0, c, false, false);
    }
#pragma unroll
    for (int v = 0; v < 8; ++v)
        h1[(size_t)(mt * 16 + v + 8 * g) * F_HID + n] = c[v];
}

// ---------------------------------------------------------------------------
// GEMM2: h2[100000x40] = relu(agg1 + b1) @ W2[32x40]. 3 N-tiles (last masked).
// Bias+ReLU fused into the A-operand load.
// ---------------------------------------------------------------------------
__global__ void gcn_gemm2(const float* __restrict__ agg1,
                          const float* __restrict__ b1,
                          const float* __restrict__ W2,
                          float* __restrict__ h2)
{
    const int wave = (blockIdx.x * blockDim.x + threadIdx.x) >> 5;
    const int lane = threadIdx.x & 31;
    if (wave >= M_TILES * 3) return;
    const int mt = wave / 3;
    const int nt = wave % 3;
    const int mi = lane & 15;
    const int g  = lane >> 4;
    const int n  = nt * 16 + mi;
    const bool nok = (n < F_OUT);
    const int  nc  = nok ? n : (F_OUT - 1);   // clamp: lanes with n>=40 compute garbage, never stored

    const float* __restrict__ arow = agg1 + (size_t)(mt * 16 + mi) * F_HID;
    v8f c = {};
#pragma unroll
    for (int kt = 0; kt < F_HID / 4; ++kt) {
        const int k0 = kt * 4 + 2 * g;
        v2f a, b;
        a.x = fmaxf(arow[k0]     + b1[k0],     0.0f);
        a.y = fmaxf(arow[k0 + 1] + b1[k0 + 1], 0.0f);
        b.x = W2[(size_t)k0 * F_OUT + nc];
        b.y = W2[(size_t)(k0 + 1) * F_OUT + nc];
        c = __builtin_amdgcn_wmma_f32_16x16x4_f32(false, a, false, b,
                                                  (short)0, c, false, false);
    }
    if (nok) {
#pragma unroll
        for (int v = 0; v < 8; ++v)
            h2[(size_t)(mt * 16 + v + 8 * g) * F_OUT + n] = c[v];
    }
}

// ---------------------------------------------------------------------------
// Edge scatter, 32 features: one wave per edge, lane f = feature.
// Gather h[src] (one 128B line), scale, native f32 atomic add into agg[dst].
// ---------------------------------------------------------------------------
__global__ void gcn_scatter32(const float* __restrict__ h,
                              const long long* __restrict__ src,
                              const long long* __restrict__ dst,
                              const float* __restrict__ ew,
                              float* __restrict__ agg)
{
    const int t = blockIdx.x * blockDim.x + threadIdx.x;
    const int e = t >> 5;
    if (e >= N_EDGES) return;
    const int f = t & 31;
    const long long s = src[e];
    const long long d = dst[e];
    const float     w = ew[e];
    const float msg = h[(size_t)s * F_HID + f] * w;
    unsafeAtomicAdd(&agg[(size_t)d * F_HID + f], msg);
}

// Edge scatter, 40 features: lane covers f=lane and (lane<8) f=lane+32.
__global__ void gcn_scatter40(const float* __restrict__ h,
                              const long long* __restrict__ src,
                              const long long* __restrict__ dst,
                              const float* __restrict__ ew,
                              float* __restrict__ agg)
{
    const int t = blockIdx.x * blockDim.x + threadIdx.x;
    const int e = t >> 5;
    if (e >= N_EDGES) return;
    const int lane = t & 31;
    const long long s = src[e];
    const long long d = dst[e];
    const float     w = ew[e];
    const float* __restrict__ hr = h   + (size_t)s * F_OUT;
    float* __restrict__       ar = agg + (size_t)d * F_OUT;
    unsafeAtomicAdd(&ar[lane], hr[lane] * w);
    if (lane < F_OUT - 32)
        unsafeAtomicAdd(&ar[32 + lane], hr[32 + lane] * w);
}

// ---------------------------------------------------------------------------
// Bias + log_softmax over 40 classes, one wave per row (wave32 shuffles).
// ---------------------------------------------------------------------------
__global__ void gcn_logsoftmax(const float* __restrict__ agg2,
                               const float* __restrict__ b2,
                               float* __restrict__ out)
{
    const int t = blockIdx.x * blockDim.x + threadIdx.x;
    const int row = t >> 5;
    if (row >= N_NODES) return;
    const int lane = t & 31;
    const float* __restrict__ ar = agg2 + (size_t)row * F_OUT;

    const float v0 = ar[lane] + b2[lane];
    const float v1 = (lane < 8) ? (ar[32 + lane] + b2[32 + lane]) : -3.0e38f;

    float mx = fmaxf(v0, v1);
#pragma unroll
    for (int off = 16; off > 0; off >>= 1)
        mx = fmaxf(mx, __shfl_xor(mx, off, 32));

    float se = __expf(v0 - mx) + ((lane < 8) ? __expf(v1 - mx) : 0.0f);
#pragma unroll
    for (int off = 16; off > 0; off >>= 1)
        se += __shfl_xor(se, off, 32);

    const float lse = mx + __logf(se);
    float* __restrict__ orow = out + (size_t)row * F_OUT;
    orow[lane] = v0 - lse;
    if (lane < 8) orow[32 + lane] = v1 - lse;
}

extern "C" void kernel_launch(void* const* d_in, const int* in_sizes, int n_in,
                              void* d_out, int out_size, void* d_ws, size_t ws_size,
                              hipStream_t stream) {
    (void)in_sizes; (void)n_in; (void)out_size; (void)ws_size;

    const float*     x   = (const float*)d_in[0];
    const long long* ei  = (const long long*)d_in[1];   // int64 [2, E]
    const float*     ew  = (const float*)d_in[2];
    const float*     W1  = (const float*)d_in[3];
    const float*     b1  = (const float*)d_in[4];
    const float*     W2  = (const float*)d_in[5];
    const float*     b2  = (const float*)d_in[6];
    float*           out = (float*)d_out;

    const long long* src = ei;
    const long long* dst = ei + (size_t)N_EDGES;

    // Workspace: hbuf holds h1 (stride 32) then h2 (stride 40);
    //            abuf holds agg1 then agg2. 2 * 100000*40*4B = 32 MB total.
    float* hbuf = (float*)d_ws;
    float* abuf = hbuf + (size_t)N_NODES * F_OUT;

    const int TPB = 256;  // 8 waves per block

    // Layer 1 linear: h1 = x @ W1
    {
        const int waves  = M_TILES * 2;
        const int blocks = (waves * 32 + TPB - 1) / TPB;
        gcn_gemm1<<<blocks, TPB, 0, stream>>>(x, W1, hbuf);
    }
    // agg1 = segment_sum(h1[src] * w, dst)
    hipMemsetAsync(abuf, 0, (size_t)N_NODES * F_HID * sizeof(float), stream);
    {
        const int blocks = (N_EDGES * 32) / TPB;  // exact: 400000
        gcn_scatter32<<<blocks, TPB, 0, stream>>>(hbuf, src, dst, ew, abuf);
    }
    // Layer 2 linear (fused bias+relu on A): h2 = relu(agg1 + b1) @ W2
    {
        const int waves  = M_TILES * 3;
        const int blocks = (waves * 32 + TPB - 1) / TPB;
        gcn_gemm2<<<blocks, TPB, 0, stream>>>(abuf, b1, W2, hbuf);
    }
    // agg2 = segment_sum(h2[src] * w, dst)
    hipMemsetAsync(abuf, 0, (size_t)N_NODES * F_OUT * sizeof(float), stream);
    {
        const int blocks = (N_EDGES * 32) / TPB;
        gcn_scatter40<<<blocks, TPB, 0, stream>>>(hbuf, src, dst, ew, abuf);
    }
    // out = log_softmax(agg2 + b2)
    {
        const int blocks = (N_NODES * 32 + TPB - 1) / TPB;
        gcn_logsoftmax<<<blocks, TPB, 0, stream>>>(abuf, b2, out);
    }
}